// GridGraph_17480516894971
// MI455X (gfx1250) — compile-verified
//
#include <hip/hip_runtime.h>

// GridGraph rook-adjacency COO generation for a 2048x2048 all-active grid.
// Output layout in d_out (12*N floats):
//   [ values(4N) | rows(4N) | cols(4N) ], each split into 4 direction blocks
//   of N in ROOK order: down, up, right, left.
// Pure streaming kernel: ~201 MB NT b128 stores + ~21 MB L2-cached loads.
// Roofline: ~222 MB / 23.3 TB/s ~= 9.5 us; zero FLOPs -> WMMA inapplicable.
// CDNA5 levers: b128 coalesced stores with TH=NT (keep the 192 MB L2 clean
// for the 4x-reused weights), global_prefetch_b8, wave32, branchless masks.

typedef float v4f __attribute__((ext_vector_type(4)));

constexpr int kH = 2048;
constexpr int kW = 2048;
constexpr int kN = kH * kW;                     // 4,194,304 (< 2^31, fits int)

__global__ __launch_bounds__(256) void grid_adjacency_kernel(
    const unsigned char* __restrict__ act,      // activities, 1 byte per vertex
    const float*         __restrict__ wts,      // vertex weights (f32)
    float*               __restrict__ out)      // 12*N floats
{
    // Grid exactly covers N/4 threads (4096 blocks x 256): no bounds check.
    const int n = (blockIdx.x * 256 + threadIdx.x) * 4;  // first of 4 vertices
    const int i = n >> 11;                      // n / 2048
    const int j = n & (kW - 1);                 // n % 2048

    const bool hasDn = (i + 1) < kH;
    const bool hasUp = i > 0;
    const bool hasR4 = (j + 4) < kW;            // right-neighbor of elem 3
    const bool hasL  = j > 0;                   // left-neighbor of elem 0

    // Streaming prefetch of the weight row needed two rows from now
    // (uniform branch: a 256-thread block spans half of one grid row).
    if (i + 2 < kH) __builtin_prefetch(wts + n + 2 * kW, 0, 1);

    // Branchless boundary handling: clamp the *offset* (reference clips the
    // index then masks the value, so any in-bounds read is equivalent).
    const int oDn = hasDn ? kW  : 0;
    const int oUp = hasUp ? -kW : 0;
    const int oR  = hasR4 ? 4   : 0;
    const int oL  = hasL  ? -1  : 0;

    // ---- loads: 3x b128 + 3x b32 + 2 scalar taps, all unconditional ----
    const v4f          wc  = *(const v4f*)(wts + n);
    const unsigned int ac  = *(const unsigned int*)(act + n);
    const v4f          wdn = *(const v4f*)(wts + n + oDn);
    const unsigned int adn = *(const unsigned int*)(act + n + oDn);
    const v4f          wup = *(const v4f*)(wts + n + oUp);
    const unsigned int aup = *(const unsigned int*)(act + n + oUp);
    const float         wR4 = wts[n + oR];
    const unsigned char aR4 = act[n + oR];
    const float         wL  = wts[n + oL];
    const unsigned char aL  = act[n + oL];

    // Index generation in f32: n < 2^22 and all results < 2^24 -> exact.
    const float f0 = (float)n;

    float v[4][4], r[4][4], c[4][4];            // [direction][element]
#pragma unroll
    for (int e = 0; e < 4; ++e) {
        const float fsrc = f0 + (float)e;

        // d0: down (i+1, j)
        {
            const bool valid = hasDn && ((adn & (0xffu << (8 * e))) != 0u);
            v[0][e] = valid ? wdn[e]          : 0.f;
            r[0][e] = valid ? fsrc            : 0.f;
            c[0][e] = valid ? fsrc + 2048.0f  : 0.f;
        }
        // d1: up (i-1, j)
        {
            const bool valid = hasUp && ((aup & (0xffu << (8 * e))) != 0u);
            v[1][e] = valid ? wup[e]          : 0.f;
            r[1][e] = valid ? fsrc            : 0.f;
            c[1][e] = valid ? fsrc - 2048.0f  : 0.f;
        }
        // d2: right (i, j+1)
        {
            const bool  aOk = (e < 3) ? ((ac & (0xffu << (8 * (e + 1)))) != 0u)
                                      : (hasR4 && aR4 != 0);
            const float wr  = (e < 3) ? wc[e + 1] : wR4;
            v[2][e] = aOk ? wr              : 0.f;
            r[2][e] = aOk ? fsrc            : 0.f;
            c[2][e] = aOk ? fsrc + 1.0f     : 0.f;
        }
        // d3: left (i, j-1)
        {
            const bool  aOk = (e > 0) ? ((ac & (0xffu << (8 * (e - 1)))) != 0u)
                                      : (hasL && aL != 0);
            const float wl  = (e > 0) ? wc[e - 1] : wL;
            v[3][e] = aOk ? wl              : 0.f;
            r[3][e] = aOk ? fsrc            : 0.f;
            c[3][e] = aOk ? fsrc - 1.0f     : 0.f;
        }
    }

    // ---- 12x b128 non-temporal streaming stores ----
    float* const vals = out;
    float* const rows = out + 4 * kN;
    float* const cols = out + 8 * kN;
#pragma unroll
    for (int d = 0; d < 4; ++d) {
        const int o = d * kN + n;
        v4f vv = { v[d][0], v[d][1], v[d][2], v[d][3] };
        v4f rv = { r[d][0], r[d][1], r[d][2], r[d][3] };
        v4f cv = { c[d][0], c[d][1], c[d][2], c[d][3] };
        __builtin_nontemporal_store(vv, (v4f*)(vals + o));
        __builtin_nontemporal_store(rv, (v4f*)(rows + o));
        __builtin_nontemporal_store(cv, (v4f*)(cols + o));
    }
}

extern "C" void kernel_launch(void* const* d_in, const int* in_sizes, int n_in,
                              void* d_out, int out_size, void* d_ws, size_t ws_size,
                              hipStream_t stream) {
    (void)in_sizes; (void)n_in; (void)out_size; (void)d_ws; (void)ws_size;
    const unsigned char* act = (const unsigned char*)d_in[0]; // bool, 1 B/elem
    const float*         wts = (const float*)d_in[1];         // f32 weights
    float*               dst = (float*)d_out;                 // 12*N floats

    const int block = 256;                       // 8 wave32 waves / workgroup
    const int grid  = kN / 4 / block;            // 4096, exact cover
    grid_adjacency_kernel<<<grid, block, 0, stream>>>(act, wts, dst);
}